// RnnForNeurons_4432406249838
// MI455X (gfx1250) — compile-verified
//
#include <hip/hip_runtime.h>
#include <hip/hip_bf16.h>

typedef __attribute__((ext_vector_type(2))) float v2f;
typedef __attribute__((ext_vector_type(8))) float v8f;

#define NUM_NEURONS 8192
#define IN_SZ 512
#define OUT_SZ 512
#define TAU_INV 0.1f
#define G_GAIN 1.5f

// ---------------------------------------------------------------------------
// Prep: rates = G * tanh(potentials); zero the shared accumulator and the
// output head of d_out (harness poisons them with 0xAA before timing).
// ---------------------------------------------------------------------------
__global__ void prep_kernel(const float* __restrict__ pot,
                            float* __restrict__ rates_scaled,
                            float* __restrict__ acc,
                            float* __restrict__ out_lo) {
    int n = blockIdx.x * blockDim.x + threadIdx.x;
    if (n < NUM_NEURONS) {
        rates_scaled[n] = G_GAIN * tanhf(pot[n]);
        acc[n] = 0.0f;
        if (n < OUT_SZ) out_lo[n] = 0.0f;
    }
}

// ---------------------------------------------------------------------------
// Generic f32 GEMV via V_WMMA_F32_16X16X4_F32:
//   acc[n] += sum_{k in [kStart,kStart+kChunk)} vec[k] * mat[k*N + n]
//
// One wave owns a 16-column tile. Per iteration (K step = 4):
//   A (16x4, 2 VGPRs): vector in matrix row M=0 only.
//     VGPR0: lanes0-15 K=0, lanes16-31 K=2 ; VGPR1: K=1 / K=3 (ISA 7.12.2)
//     -> lane 0 holds vec[k0],vec[k0+1]; lane 16 holds vec[k0+2],vec[k0+3];
//        all other lanes 0 (v_cndmask, no divergence -> EXEC stays all-1s).
//   B (4x16, 2 VGPRs): lo lanes carry rows k0,k0+1; hi lanes rows k0+2,k0+3,
//     column = n0 + (lane&15). Per-lane offsets are loop-invariant; only the
//     row base advances (4*N floats per iter). Nontemporal: synapses (256MB)
//     exceeds the 192MB L2, so stream it.
//   D row M=0 accumulates in c[0] lanes 0-15 -> one atomicAdd per column.
// ---------------------------------------------------------------------------
__global__ void gemv_wmma_f32(const float* __restrict__ vec,
                              const float* __restrict__ mat,
                              float* __restrict__ acc,
                              int N, int kChunk) {
    const int lane = threadIdx.x & 31;
    const int wave = threadIdx.x >> 5;
    const int tile = blockIdx.x * (blockDim.x >> 5) + wave;
    const int n0   = tile * 16;

    const int kStart = blockIdx.y * kChunk;
    const int col    = lane & 15;
    const int rowOff = (lane >> 4) << 1;      // 0 for lanes 0-15, 2 for 16-31
    const bool sel   = (col == 0);            // lanes 0 and 16 carry the vector

    const float* bptr  = mat + (size_t)(kStart + rowOff) * N + (size_t)(n0 + col);
    const float* aptr  = vec + kStart + rowOff;
    const size_t bstep = (size_t)4 * N;

    v8f c = {};
    #pragma unroll 4
    for (int k = 0; k < kChunk; k += 4) {
        v2f b;
        b.x = __builtin_nontemporal_load(bptr);
        b.y = __builtin_nontemporal_load(bptr + N);
        float a0 = aptr[0];
        float a1 = aptr[1];
        v2f a;
        a.x = sel ? a0 : 0.0f;
        a.y = sel ? a1 : 0.0f;
        // (neg_a, A, neg_b, B, c_mod, C, reuse_a, reuse_b)
        c = __builtin_amdgcn_wmma_f32_16x16x4_f32(false, a, false, b,
                                                  (short)0, c, false, false);
        bptr += bstep;
        aptr += 4;
    }

    if (lane < 16) {
        atomicAdd(&acc[n0 + lane], c[0]);     // global_atomic_add_f32
    }
}

// ---------------------------------------------------------------------------
// Leaky-integrator update: pot_new = pot + (acc - pot)/tau ; gen = tanh(pot_new)
// acc already holds G*(tanh(pot)@synapses) + inputs@W_in.
// ---------------------------------------------------------------------------
__global__ void pointwise_kernel(const float* __restrict__ pot,
                                 const float* __restrict__ acc,
                                 float* __restrict__ pot_out,
                                 float* __restrict__ gen) {
    int n = blockIdx.x * blockDim.x + threadIdx.x;
    if (n < NUM_NEURONS) {
        float p  = pot[n];
        float pn = p + (acc[n] - p) * TAU_INV;
        pot_out[n] = pn;
        gen[n] = tanhf(pn);
    }
}

extern "C" void kernel_launch(void* const* d_in, const int* in_sizes, int n_in,
                              void* d_out, int out_size, void* d_ws, size_t ws_size,
                              hipStream_t stream) {
    const float* inputs   = (const float*)d_in[0];   // [512]
    const float* pot      = (const float*)d_in[1];   // [1, 8192]
    const float* W_in     = (const float*)d_in[2];   // [512, 8192]
    const float* synapses = (const float*)d_in[3];   // [8192, 8192]
    const float* W_out    = (const float*)d_in[4];   // [8192, 512]
    float* out = (float*)d_out;                      // [512 output | 8192 pot_new]

    float* ws    = (float*)d_ws;
    float* rates = ws;                 // 8192 floats: G*tanh(pot)
    float* acc   = ws + NUM_NEURONS;   // 8192 floats: recurrent+input drive
    float* gen   = ws + 2 * NUM_NEURONS; // 8192 floats: tanh(pot_new)

    // 1) rates, zero acc, zero out[0:512]
    prep_kernel<<<dim3(NUM_NEURONS / 256), dim3(256), 0, stream>>>(pot, rates, acc, out);

    // 2) acc += rates @ synapses   (K=8192, N=8192; 8-way K-split -> 4096 waves)
    gemv_wmma_f32<<<dim3(NUM_NEURONS / 64, 8), dim3(128), 0, stream>>>(
        rates, synapses, acc, NUM_NEURONS, 1024);

    // 3) acc += inputs @ W_in      (K=512, N=8192)
    gemv_wmma_f32<<<dim3(NUM_NEURONS / 64, 1), dim3(128), 0, stream>>>(
        inputs, W_in, acc, NUM_NEURONS, 512);

    // 4) pot_new -> d_out[512:], gen = tanh(pot_new)
    pointwise_kernel<<<dim3(NUM_NEURONS / 256), dim3(256), 0, stream>>>(
        pot, acc, out + OUT_SZ, gen);

    // 5) out[0:512] += gen @ W_out (K=8192, N=512; 16-way K-split)
    gemv_wmma_f32<<<dim3(OUT_SZ / 64, 16), dim3(128), 0, stream>>>(
        gen, W_out, out, OUT_SZ, 512);
}